// GraphVAMPNetLayer_61400852463923
// MI455X (gfx1250) — compile-verified
//
#include <hip/hip_runtime.h>

typedef float v2f __attribute__((ext_vector_type(2)));
typedef float v8f __attribute__((ext_vector_type(8)));

#define WMMA_F32X4(a, b, c) \
  __builtin_amdgcn_wmma_f32_16x16x4_f32(false, (a), false, (b), (short)0, (c), false, false)

namespace {
constexpr int NUM_ATOMS  = 100;
constexpr int NUM_GRAPHS = 1000;
constexpr int NN   = NUM_ATOMS * NUM_GRAPHS;   // 100000 nodes
constexpr int DEG  = 20;
constexpr int H    = 32;
constexpr int ED   = 51;                       // RBF centers
constexpr int NTILES = NN / 16;                // 6250 16-row tiles
constexpr float STEP = 0.2f;
constexpr float INV_STEP2 = 25.0f;             // 1/STEP^2
}

// ---------------------------------------------------------------------------
// atom_emb[n][c] = emb_table[n % 100][c]
__global__ __launch_bounds__(256) void initEmbKernel(
    const float* __restrict__ embTable, float* __restrict__ atomEmb)
{
  int idx = blockIdx.x * 256 + threadIdx.x;
  if (idx < NN * H) {
    int n = idx >> 5, c = idx & 31;
    atomEmb[idx] = embTable[(n % NUM_ATOMS) * H + c];
  }
}

// ---------------------------------------------------------------------------
// h = atom_emb @ W^T   (NN x 32) @ (32 x 32), one wave per 16-row tile.
__global__ __launch_bounds__(32) void nodeHKernel(
    const float* __restrict__ emb, const float* __restrict__ W,
    float* __restrict__ h)
{
  const int lane  = threadIdx.x & 31;
  const int mrow  = lane & 15;
  const int khalf = lane >> 4;
  const int tile  = blockIdx.x;
  const int row   = tile * 16 + mrow;

  v8f c0 = {};
  v8f c1 = {};
#pragma unroll
  for (int kk = 0; kk < 8; ++kk) {
    const int kb = 4 * kk + 2 * khalf;
    v2f a  = *(const v2f*)&emb[row * H + kb];
    v2f b0 = *(const v2f*)&W[(mrow)      * H + kb];   // B[k][n] = W[n][k]
    v2f b1 = *(const v2f*)&W[(mrow + 16) * H + kb];
    c0 = WMMA_F32X4(a, b0, c0);
    c1 = WMMA_F32X4(a, b1, c1);
  }
#pragma unroll
  for (int r = 0; r < 8; ++r) {
    const int m = tile * 16 + r + 8 * khalf;
    h[m * H + mrow]      = c0[r];
    h[m * H + mrow + 16] = c1[r];
  }
}

// ---------------------------------------------------------------------------
// Edge + attention + contiguous segment reduction.
// One WG = 160 threads (5 waves) owns 4 nodes = 80 consecutive edges
// = 5 WMMA tiles of 16 edges.  No atomics: src = edge/20 is contiguous.
__global__ __launch_bounds__(160) void edgeConvKernel(
    const float* __restrict__ data,  const float* __restrict__ h,
    const float* __restrict__ Wf1,   const float* __restrict__ bf1,
    const float* __restrict__ Wf2,   const float* __restrict__ bf2,
    const float* __restrict__ coef,  float* __restrict__ conv)
{
  __shared__ __align__(16) float sWf1[32][52];   // padded, col 51 = 0
  __shared__ __align__(16) float sWf2[32][32];
  __shared__ float sbf1[32], sbf2[32], scoef[32];
  __shared__ __align__(16) float sT[5][16][34];  // per-wave transpose scratch
  __shared__ __align__(16) float sMsg[80][33];
  __shared__ float sAttn[80];
  __shared__ float sDenom[4];

  const int t     = threadIdx.x;
  const int w     = t >> 5;
  const int lane  = t & 31;
  const int mrow  = lane & 15;
  const int khalf = lane >> 4;
  const int nodeBase = blockIdx.x * 4;
  const int edgeBase = blockIdx.x * 80;

  // --- stage per-layer weights into LDS -----------------------------------
  for (int idx = t; idx < 32 * 52; idx += 160) {
    int r = idx / 52, c = idx % 52;
    sWf1[r][c] = (c < ED) ? Wf1[r * ED + c] : 0.0f;
  }
  for (int idx = t; idx < 32 * 32; idx += 160)
    sWf2[idx >> 5][idx & 31] = Wf2[idx];
  if (t < 32) { sbf1[t] = bf1[t]; sbf2[t] = bf2[t]; scoef[t] = coef[t]; }
  __syncthreads();

  // --- GEMM1: rbf(d) (16x52) @ Wf1^T (52x32) ------------------------------
  const int tileBase = edgeBase + w * 16;
  const float dm = data[3 * (tileBase + mrow) + 2];

  v2f aArr[13];
#pragma unroll
  for (int kk = 0; kk < 13; ++kk) {
    const int kb = 4 * kk + 2 * khalf;
    float x0 = dm - STEP * (float)kb;
    float a0 = expf(-x0 * x0 * INV_STEP2);
    float a1 = 0.0f;
    if (kb + 1 <= ED - 1) {
      float x1 = dm - STEP * (float)(kb + 1);
      a1 = expf(-x1 * x1 * INV_STEP2);
    }
    v2f a; a.x = a0; a.y = a1;
    aArr[kk] = a;
  }

  v8f c0 = {};
  v8f c1 = {};
#pragma unroll
  for (int kk = 0; kk < 13; ++kk) {
    const int kb = 4 * kk + 2 * khalf;
    v2f b0 = *(const v2f*)&sWf1[mrow][kb];
    v2f b1 = *(const v2f*)&sWf1[mrow + 16][kb];
    c0 = WMMA_F32X4(aArr[kk], b0, c0);
    c1 = WMMA_F32X4(aArr[kk], b1, c1);
  }

  // tanh(+bf1), transpose to LDS (D-layout -> A-layout)
#pragma unroll
  for (int r = 0; r < 8; ++r) {
    const int m = r + 8 * khalf;
    sT[w][m][mrow]      = tanhf(c0[r] + sbf1[mrow]);
    sT[w][m][mrow + 16] = tanhf(c1[r] + sbf1[mrow + 16]);
  }
  __syncthreads();

  // --- GEMM2: T (16x32) @ Wf2^T (32x32), C preloaded with bf2 -------------
  v8f d0, d1;
  {
    const float b2a = sbf2[mrow], b2b = sbf2[mrow + 16];
#pragma unroll
    for (int r = 0; r < 8; ++r) { d0[r] = b2a; d1[r] = b2b; }
  }
#pragma unroll
  for (int kk = 0; kk < 8; ++kk) {
    const int kb = 4 * kk + 2 * khalf;
    v2f a  = *(const v2f*)&sT[w][mrow][kb];
    v2f b0 = *(const v2f*)&sWf2[mrow][kb];
    v2f b1 = *(const v2f*)&sWf2[mrow + 16][kb];
    d0 = WMMA_F32X4(a, b0, d0);
    d1 = WMMA_F32X4(a, b1, d1);
  }

  // --- msg = h[dst] * he, stash in LDS ------------------------------------
#pragma unroll
  for (int r = 0; r < 8; ++r) {
    const int m   = r + 8 * khalf;
    const int e   = tileBase + m;
    const int dst = (int)data[3 * e + 1];
    sMsg[w * 16 + m][mrow]      = d0[r] * h[dst * H + mrow];
    sMsg[w * 16 + m][mrow + 16] = d1[r] * h[dst * H + mrow + 16];
  }
  __syncthreads();

  // --- attn = exp(msg . coef) ---------------------------------------------
  if (t < 80) {
    float s = 0.0f;
#pragma unroll
    for (int c = 0; c < H; ++c) s += sMsg[t][c] * scoef[c];
    sAttn[t] = expf(s);
  }
  __syncthreads();
  if (t < 4) {
    float dsum = 0.0f;
#pragma unroll
    for (int k = 0; k < DEG; ++k) dsum += sAttn[t * DEG + k];
    sDenom[t] = dsum;
  }
  __syncthreads();

  // --- contiguous softmax-weighted reduction, write conv ------------------
  if (t < 128) {
    const int ch = t & 31, j = t >> 5;
    float num = 0.0f;
#pragma unroll
    for (int k = 0; k < DEG; ++k)
      num += sMsg[j * DEG + k][ch] * sAttn[j * DEG + k];
    const float den = sDenom[j];
    conv[(nodeBase + j) * H + ch] = (den > 0.0f) ? num / den : num;
  }
}

// ---------------------------------------------------------------------------
// out = tanh(conv @ Wo1^T + bo1) @ Wo2^T + bo2 ; atom_emb += out
__global__ __launch_bounds__(32) void nodeOutKernel(
    const float* __restrict__ conv,
    const float* __restrict__ Wo1, const float* __restrict__ bo1,
    const float* __restrict__ Wo2, const float* __restrict__ bo2,
    float* __restrict__ emb)
{
  __shared__ __align__(16) float sT[16][34];

  const int lane  = threadIdx.x & 31;
  const int mrow  = lane & 15;
  const int khalf = lane >> 4;
  const int tile  = blockIdx.x;
  const int row   = tile * 16 + mrow;

  v8f c0, c1;
  {
    const float ba = bo1[mrow], bb = bo1[mrow + 16];
#pragma unroll
    for (int r = 0; r < 8; ++r) { c0[r] = ba; c1[r] = bb; }
  }
#pragma unroll
  for (int kk = 0; kk < 8; ++kk) {
    const int kb = 4 * kk + 2 * khalf;
    v2f a  = *(const v2f*)&conv[row * H + kb];
    v2f b0 = *(const v2f*)&Wo1[(mrow)      * H + kb];
    v2f b1 = *(const v2f*)&Wo1[(mrow + 16) * H + kb];
    c0 = WMMA_F32X4(a, b0, c0);
    c1 = WMMA_F32X4(a, b1, c1);
  }
#pragma unroll
  for (int r = 0; r < 8; ++r) {
    const int m = r + 8 * khalf;
    sT[m][mrow]      = tanhf(c0[r]);
    sT[m][mrow + 16] = tanhf(c1[r]);
  }
  __syncthreads();

  v8f d0, d1;
  {
    const float ba = bo2[mrow], bb = bo2[mrow + 16];
#pragma unroll
    for (int r = 0; r < 8; ++r) { d0[r] = ba; d1[r] = bb; }
  }
#pragma unroll
  for (int kk = 0; kk < 8; ++kk) {
    const int kb = 4 * kk + 2 * khalf;
    v2f a  = *(const v2f*)&sT[mrow][kb];
    v2f b0 = *(const v2f*)&Wo2[(mrow)      * H + kb];
    v2f b1 = *(const v2f*)&Wo2[(mrow + 16) * H + kb];
    d0 = WMMA_F32X4(a, b0, d0);
    d1 = WMMA_F32X4(a, b1, d1);
  }
#pragma unroll
  for (int r = 0; r < 8; ++r) {
    const int m = tile * 16 + r + 8 * khalf;
    emb[m * H + mrow]      += d0[r];
    emb[m * H + mrow + 16] += d1[r];
  }
}

// ---------------------------------------------------------------------------
// pooled = mean(relu(emb)) per graph ; out = pooled @ Wg^T + bg  (1000 x 16)
__global__ __launch_bounds__(32) void poolKernel(
    const float* __restrict__ emb, const float* __restrict__ Wg,
    const float* __restrict__ bg, float* __restrict__ out)
{
  __shared__ float sp[32];
  const int g = blockIdx.x;
  const int lane = threadIdx.x;
  const float* base = emb + (size_t)g * NUM_ATOMS * H;

  float s = 0.0f;
  for (int a = 0; a < NUM_ATOMS; ++a)
    s += fmaxf(base[a * H + lane], 0.0f);
  sp[lane] = s * (1.0f / (float)NUM_ATOMS);
  __syncthreads();

  if (lane < 16) {
    float o = bg[lane];
#pragma unroll
    for (int c = 0; c < H; ++c) o += sp[c] * Wg[lane * H + c];
    out[g * 16 + lane] = o;
  }
}

// ---------------------------------------------------------------------------
extern "C" void kernel_launch(void* const* d_in, const int* in_sizes, int n_in,
                              void* d_out, int out_size, void* d_ws, size_t ws_size,
                              hipStream_t stream) {
  (void)in_sizes; (void)n_in; (void)out_size; (void)ws_size;
  const float* data  = (const float*)d_in[0];
  const float* embT  = (const float*)d_in[1];
  const float* Winit = (const float*)d_in[2];
  const float* Wf1   = (const float*)d_in[3];
  const float* bf1   = (const float*)d_in[4];
  const float* Wf2   = (const float*)d_in[5];
  const float* bf2   = (const float*)d_in[6];
  const float* coef  = (const float*)d_in[7];
  const float* Wo1   = (const float*)d_in[8];
  const float* bo1   = (const float*)d_in[9];
  const float* Wo2   = (const float*)d_in[10];
  const float* bo2   = (const float*)d_in[11];
  const float* Wg    = (const float*)d_in[12];
  const float* bg    = (const float*)d_in[13];
  float* out = (float*)d_out;

  float* ws      = (float*)d_ws;
  float* atomEmb = ws;
  float* hBuf    = ws + (size_t)NN * H;
  float* convBuf = ws + (size_t)2 * NN * H;

  initEmbKernel<<<(NN * H + 255) / 256, 256, 0, stream>>>(embT, atomEmb);

  for (int i = 0; i < 3; ++i) {
    nodeHKernel<<<NTILES, 32, 0, stream>>>(atomEmb, Winit + i * H * H, hBuf);
    edgeConvKernel<<<NN / 4, 160, 0, stream>>>(
        data, hBuf,
        Wf1 + i * H * ED, bf1 + i * H,
        Wf2 + i * H * H,  bf2 + i * H,
        coef + i * H, convBuf);
    nodeOutKernel<<<NTILES, 32, 0, stream>>>(
        convBuf, Wo1 + i * H * H, bo1 + i * H,
        Wo2 + i * H * H, bo2 + i * H, atomEmb);
  }

  poolKernel<<<NUM_GRAPHS, 32, 0, stream>>>(atomEmb, Wg, bg, out);
}